// MarginLoss_83141976916760
// MI455X (gfx1250) — compile-verified
//
#include <hip/hip_runtime.h>

// MarginLoss over N=2^26 f32 elements for MI455X (gfx1250, wave32).
// Memory-bound: ~512MB traffic -> ~22us at 23.3 TB/s. Streams with NT b128
// loads; wave reductions use V_WMMA_F32_16X16X4_F32 (exact f32 matrix math).

typedef __attribute__((ext_vector_type(4))) float v4f;
typedef __attribute__((ext_vector_type(4))) int   v4i;
typedef __attribute__((ext_vector_type(2))) float v2f;
typedef __attribute__((ext_vector_type(8))) float v8f;

#define MARGIN 1.0f
#define EPS    1e-6f
#define NB     8192
#define TPB    256

// Sum x over all 32 lanes of the wave using the matrix pipe.
// A (16x4 f32) = per-lane sums in VGPR0, zeros in VGPR1; B = ones.
// D[m][n] = A[m][0] + A[m][2] = sum of lane m and lane m+16 contributions.
// Summing a lane's 8 D VGPRs gives: lanes 0-15 -> sum(rows 0..7),
// lanes 16-31 -> sum(rows 8..15); one shfl_xor(16) completes the total.
static __device__ __forceinline__ float wave_sum_wmma(float x) {
    v2f a; a[0] = x;    a[1] = 0.0f;
    v2f b; b[0] = 1.0f; b[1] = 1.0f;
    v8f c = {};
    v8f d = __builtin_amdgcn_wmma_f32_16x16x4_f32(
        /*neg_a=*/false, a, /*neg_b=*/false, b,
        /*c_mod=*/(short)0, c, /*reuse_a=*/false, /*reuse_b=*/false);
    float s = ((d[0] + d[1]) + (d[2] + d[3])) + ((d[4] + d[5]) + (d[6] + d[7]));
    s += __shfl_xor(s, 16);
    return s;  // full wave sum in every lane
}

// Phase 0: argmax fallback (all-zero labels -> index 0), single thread.
__global__ void ml_init_kernel(const float* __restrict__ logits,
                               float* __restrict__ max_score) {
    *max_score = logits[0];
}

// Phase 1: stream labels (one-hot) as b128 NT loads; unique writer stores
// the matching logit as max_score.
__global__ void ml_find_kernel(const v4i* __restrict__ labels4,
                               const float* __restrict__ logits,
                               float* __restrict__ max_score,
                               int n4, int n) {
    int tid    = blockIdx.x * blockDim.x + threadIdx.x;
    int stride = gridDim.x * blockDim.x;
    for (int i = tid; i < n4; i += stride) {
        v4i v = __builtin_nontemporal_load(&labels4[i]);
        if ((v[0] | v[1] | v[2] | v[3]) != 0) {
            int base = i << 2;
            if      (v[0] == 1) *max_score = logits[base + 0];
            else if (v[1] == 1) *max_score = logits[base + 1];
            else if (v[2] == 1) *max_score = logits[base + 2];
            else if (v[3] == 1) *max_score = logits[base + 3];
        }
    }
    // scalar tail (n not multiple of 4) -- handled by one thread
    if (tid == 0) {
        const int* l = (const int*)labels4;
        for (int i = n4 << 2; i < n; ++i)
            if (l[i] == 1) *max_score = logits[i];
    }
}

// Phase 2: stream logits as b128 NT loads, clamp, per-block partial sums.
__global__ void ml_clamp_sum_kernel(const v4f* __restrict__ logits4,
                                    const float* __restrict__ max_score,
                                    float* __restrict__ partial,
                                    int n4, int n) {
    const float bias = MARGIN - *max_score;  // uniform -> scalar load
    int tid    = blockIdx.x * blockDim.x + threadIdx.x;
    int stride = gridDim.x * blockDim.x;

    v4f acc; acc[0] = 0.0f; acc[1] = 0.0f; acc[2] = 0.0f; acc[3] = 0.0f;
    for (int i = tid; i < n4; i += stride) {
        v4f x = __builtin_nontemporal_load(&logits4[i]);
        acc[0] += fmaxf(x[0] + bias, EPS);
        acc[1] += fmaxf(x[1] + bias, EPS);
        acc[2] += fmaxf(x[2] + bias, EPS);
        acc[3] += fmaxf(x[3] + bias, EPS);
    }
    float t = (acc[0] + acc[1]) + (acc[2] + acc[3]);
    if (tid == 0) {  // scalar tail
        const float* lf = (const float*)logits4;
        for (int i = n4 << 2; i < n; ++i) t += fmaxf(lf[i] + bias, EPS);
    }

    // wave reduction via WMMA (EXEC all ones here), then block via LDS
    float ws = wave_sum_wmma(t);
    __shared__ float wsum[TPB / 32];
    int wave = threadIdx.x >> 5, lane = threadIdx.x & 31;
    if (lane == 0) wsum[wave] = ws;
    __syncthreads();
    if (threadIdx.x == 0) {
        float b = 0.0f;
        #pragma unroll
        for (int i = 0; i < TPB / 32; ++i) b += wsum[i];
        partial[blockIdx.x] = b;
    }
}

// Phase 3: single block reduces NB partials, writes mean.
__global__ void ml_final_kernel(const float* __restrict__ partial,
                                float* __restrict__ out, int nb, int n) {
    float t = 0.0f;
    for (int i = threadIdx.x; i < nb; i += blockDim.x) t += partial[i];

    float ws = wave_sum_wmma(t);
    __shared__ float wsum[TPB / 32];
    int wave = threadIdx.x >> 5, lane = threadIdx.x & 31;
    if (lane == 0) wsum[wave] = ws;
    __syncthreads();
    if (threadIdx.x == 0) {
        float b = 0.0f;
        #pragma unroll
        for (int i = 0; i < TPB / 32; ++i) b += wsum[i];
        out[0] = b * (1.0f / (float)n);
    }
}

extern "C" void kernel_launch(void* const* d_in, const int* in_sizes, int n_in,
                              void* d_out, int out_size, void* d_ws, size_t ws_size,
                              hipStream_t stream) {
    const float* logits = (const float*)d_in[0];
    const int*   labels = (const int*)d_in[1];
    float*       out    = (float*)d_out;
    int n  = in_sizes[0];
    int n4 = n >> 2;

    float* wsf       = (float*)d_ws;
    float* max_score = wsf;        // 1 float
    float* partial   = wsf + 16;   // NB floats (64B-offset from max_score)

    ml_init_kernel<<<1, 1, 0, stream>>>(logits, max_score);
    ml_find_kernel<<<NB, TPB, 0, stream>>>((const v4i*)labels, logits,
                                           max_score, n4, n);
    ml_clamp_sum_kernel<<<NB, TPB, 0, stream>>>((const v4f*)logits, max_score,
                                                partial, n4, n);
    ml_final_kernel<<<1, TPB, 0, stream>>>(partial, out, NB, n);
}